// DifferentiableVectorization_21268678050251
// MI455X (gfx1250) — compile-verified
//
#include <hip/hip_runtime.h>
#include <hip/hip_bf16.h>
#include <math.h>

// ---------------------------------------------------------------------------
// CDNA5 WMMA f32 16x16x4 (wave32):
//   A fragment: 16x4 fp32  -> 2 VGPRs/lane  (v2f)
//   B fragment: 4x16 fp32  -> 2 VGPRs/lane  (v2f)
//   C/D:        16x16 fp32 -> 8 VGPRs/lane  (v8f)
// ---------------------------------------------------------------------------
typedef __attribute__((ext_vector_type(2))) float v2f;
typedef __attribute__((ext_vector_type(8))) float v8f;
typedef __attribute__((ext_vector_type(4))) unsigned int u32x4;
typedef __attribute__((ext_vector_type(4))) int i32x4;
typedef __attribute__((ext_vector_type(8))) int i32x8;

#define BB   16
#define CC   256
#define PP   20
#define KK   50
#define NPTS 1000
#define H2W  128
#define FEAT_LD 260        // 258 padded to multiple of 4

#if defined(__has_builtin)
#if __has_builtin(__builtin_amdgcn_tensor_load_to_lds)
#define DV_HAVE_TDM 1
#endif
#endif
#ifndef DV_HAVE_TDM
#define DV_HAVE_TDM 0
#endif

#define DV_WMMA(a, b, c) \
    __builtin_amdgcn_wmma_f32_16x16x4_f32(false, (a), false, (b), (short)0, (c), false, false)

// ---------------------------------------------------------------------------
// TDM 2D tile load: W[rows x k_elems] (fp32, row stride ld_elems) -> LDS@0.
// Descriptor per CDNA5 ISA §8.3/8.4. Optional LDS padding of 2 dwords per
// 128 stored dwords (pad_en) so an LDS row stride of 128 becomes 130
// (bank-conflict-free, 8B aligned).
// ---------------------------------------------------------------------------
#if DV_HAVE_TDM
__device__ __forceinline__ void dv_tdm_load_2d(const float* gsrc, int k_elems,
                                               int rows, int ld_elems, int pad_en)
{
    unsigned long long ga = (unsigned long long)(const void*)gsrc;
    u32x4 g0;
    g0[0] = 1u;                                        // count=1, user D#
    g0[1] = 0u;                                        // lds_addr = 0 (tile is only LDS object)
    g0[2] = (unsigned)(ga & 0xffffffffu);              // global_addr[31:0]
    g0[3] = (unsigned)((ga >> 32) & 0x01ffffffu)       // global_addr[56:32]
          | (2u << 30);                                // type = 2 ("image")

    unsigned d0 = (2u << 16);                          // data_size = 4B
    if (pad_en)
        d0 |= (1u << 20) | (6u << 22) | (1u << 25);    // pad 2 dwords / 128 dwords

    i32x8 g1;
    g1[0] = (int)d0;
    g1[1] = (int)(((unsigned)k_elems & 0xffffu) << 16);             // tensor_dim0[15:0]
    g1[2] = (int)((((unsigned)k_elems >> 16) & 0xffffu)             // tensor_dim0[31:16]
          | (((unsigned)rows & 0xffffu) << 16));                    // tensor_dim1[15:0]
    g1[3] = (int)(((unsigned)k_elems & 0xffffu) << 16);             // tile_dim0 = k_elems
    g1[4] = (int)((unsigned)rows & 0xffffu);                        // tile_dim1 = rows, tile_dim2 = 0
    g1[5] = (int)(unsigned)ld_elems;                                // tensor_dim0_stride[31:0]
    g1[6] = 0;
    g1[7] = 0;

    i32x4 z4 = {0, 0, 0, 0};
#if __clang_major__ >= 23
    i32x8 z8 = {0, 0, 0, 0, 0, 0, 0, 0};
    __builtin_amdgcn_tensor_load_to_lds(g0, g1, z4, z4, z8, 0);
#else
    __builtin_amdgcn_tensor_load_to_lds(g0, g1, z4, z4, 0);
#endif
}
#endif

// ---------------------------------------------------------------------------
// Streaming fp32 WMMA GEMM (for long-K, few-tile layers; weights single-use
// -> pure HBM stream, K split across WAVES with LDS reduction).
// Grid: (N/16, M/16); block: (32, WAVES).
// ---------------------------------------------------------------------------
template <int WAVES>
__global__ __launch_bounds__(32 * WAVES)
void dv_wmma_gemm_stream(const float* __restrict__ A, int lda,
                         const float* __restrict__ W, int ldw,
                         const float* __restrict__ bias,
                         float* __restrict__ out, int ldo,
                         int K, int act)
{
    __shared__ float red[WAVES * 256];

    const int lane = threadIdx.x;
    const int wave = threadIdx.y;
    const int n0 = blockIdx.x << 4;
    const int m0 = blockIdx.y << 4;

    const int half = lane >> 4;
    const int l16  = lane & 15;
    const int ksel = half << 1;

    const float* __restrict__ Arow = A + (size_t)(m0 + l16) * (size_t)lda;
    const float* __restrict__ Wrow = W + (size_t)(n0 + l16) * (size_t)ldw;

    const int kSteps = (K + 3) >> 2;
    const int perW   = (kSteps + WAVES - 1) / WAVES;
    const int kBeg   = wave * perW * 4;
    int kEnd = kBeg + perW * 4;
    if (kEnd > K) kEnd = K;

    v8f acc0 = {}, acc1 = {};
    int k = kBeg;
    // 2-accumulator unroll: breaks the D->C dependency chain.
    for (; k + 8 <= kEnd; k += 8) {
        const v2f a0 = *(const v2f*)(Arow + k + ksel);
        const v2f b0 = *(const v2f*)(Wrow + k + ksel);
        const v2f a1 = *(const v2f*)(Arow + k + 4 + ksel);
        const v2f b1 = *(const v2f*)(Wrow + k + 4 + ksel);
        acc0 = DV_WMMA(a0, b0, acc0);
        acc1 = DV_WMMA(a1, b1, acc1);
    }
    for (; k + 4 <= kEnd; k += 4) {
        const v2f a = *(const v2f*)(Arow + k + ksel);
        const v2f b = *(const v2f*)(Wrow + k + ksel);
        acc0 = DV_WMMA(a, b, acc0);
    }
    if (k < kEnd) {   // K % 4 tail, zero-padded
        const int k0 = k + ksel, k1 = k0 + 1;
        v2f a, b;
        a.x = (k0 < kEnd) ? Arow[k0] : 0.0f;
        a.y = (k1 < kEnd) ? Arow[k1] : 0.0f;
        b.x = (k0 < kEnd) ? Wrow[k0] : 0.0f;
        b.y = (k1 < kEnd) ? Wrow[k1] : 0.0f;
        acc0 = DV_WMMA(a, b, acc0);
    }
    acc0 += acc1;

    #pragma unroll
    for (int r = 0; r < 8; ++r)
        red[(wave * 8 + r) * 32 + lane] = acc0[r];
    __syncthreads();
    if (wave != 0) return;

    #pragma unroll
    for (int r = 0; r < 8; ++r) {
        float s = acc0[r];
        #pragma unroll
        for (int w = 1; w < WAVES; ++w)
            s += red[(w * 8 + r) * 32 + lane];
        acc0[r] = s;
    }

    const float bn = bias[n0 + l16];
    #pragma unroll
    for (int r = 0; r < 8; ++r) {
        const int m = m0 + r + (half << 3);
        float v = acc0[r] + bn;
        if (act == 1) v = fmaxf(v, 0.0f);
        else if (act == 2) v = 1.0f / (1.0f + __expf(-v));
        out[(size_t)m * (size_t)ldo + (n0 + l16)] = v;
    }
}

// ---------------------------------------------------------------------------
// LDS-staged fp32 WMMA GEMM (short-K, many-M layers).
// Grid: (N/16, M/(16*MW)); block: (32, MW). Each wave owns one M-tile; the
// shared 16xK weight tile is DMA'd into LDS once by the TDM (wave 0), then
// B fragments come from LDS (ds_load_b64). K <= 260.
// ---------------------------------------------------------------------------
template <int MW>
__global__ __launch_bounds__(32 * MW)
void dv_wmma_gemm_lds(const float* __restrict__ A, int lda,
                      const float* __restrict__ W, int ldw,
                      const float* __restrict__ bias,
                      float* __restrict__ out, int ldo,
                      int K, int act)
{
    __shared__ __attribute__((aligned(16))) float tileW[16 * 264];

    const int lane = threadIdx.x;
    const int wave = threadIdx.y;
    const int n0 = blockIdx.x << 4;
    const int m0 = (blockIdx.y * MW + wave) << 4;

    const int half = lane >> 4;
    const int l16  = lane & 15;
    const int ksel = half << 1;

    const int padEn = ((K & 127) == 0) ? 1 : 0;            // K multiple of 128
    const int ldsK  = K + (padEn ? ((K >> 7) << 1) : 0);   // +2 dwords / 128

#if DV_HAVE_TDM
    if (wave == 0) {
        dv_tdm_load_2d(W + (size_t)n0 * (size_t)ldw, K, 16, ldw, padEn);
        __builtin_amdgcn_s_wait_tensorcnt(0);
    }
    __syncthreads();
#else
    for (int r = wave; r < 16; r += MW)
        for (int c = lane; c < K; c += 32)
            tileW[r * ldsK + c] = W[(size_t)(n0 + r) * (size_t)ldw + c];
    __syncthreads();
#endif

    const float* __restrict__ Arow = A + (size_t)(m0 + l16) * (size_t)lda;
    const float* __restrict__ Brow = &tileW[l16 * ldsK];

    v8f acc0 = {}, acc1 = {};
    int k = 0;
    for (; k + 8 <= K; k += 8) {
        const v2f a0 = *(const v2f*)(Arow + k + ksel);
        const v2f b0 = *(const v2f*)(Brow + k + ksel);
        const v2f a1 = *(const v2f*)(Arow + k + 4 + ksel);
        const v2f b1 = *(const v2f*)(Brow + k + 4 + ksel);
        acc0 = DV_WMMA(a0, b0, acc0);
        acc1 = DV_WMMA(a1, b1, acc1);
    }
    for (; k + 4 <= K; k += 4) {
        const v2f a = *(const v2f*)(Arow + k + ksel);
        const v2f b = *(const v2f*)(Brow + k + ksel);
        acc0 = DV_WMMA(a, b, acc0);
    }
    if (k < K) {      // K % 4 tail
        const int k0 = k + ksel, k1 = k0 + 1;
        v2f a, b;
        a.x = (k0 < K) ? Arow[k0] : 0.0f;
        a.y = (k1 < K) ? Arow[k1] : 0.0f;
        b.x = (k0 < K) ? Brow[k0] : 0.0f;
        b.y = (k1 < K) ? Brow[k1] : 0.0f;
        acc0 = DV_WMMA(a, b, acc0);
    }
    acc0 += acc1;

    const float bn = bias[n0 + l16];
    #pragma unroll
    for (int r = 0; r < 8; ++r) {
        const int m = m0 + r + (half << 3);
        float v = acc0[r] + bn;
        if (act == 1) v = fmaxf(v, 0.0f);
        else if (act == 2) v = 1.0f / (1.0f + __expf(-v));
        out[(size_t)m * (size_t)ldo + (n0 + l16)] = v;
    }
}

// ---------------------------------------------------------------------------
// 4x4 average pooling: p4[16,256,32,32] -> X1[16, 256*64]
// ---------------------------------------------------------------------------
__global__ void dv_pool_kernel(const float* __restrict__ p4,
                               float* __restrict__ X1)
{
    int idx = blockIdx.x * blockDim.x + threadIdx.x;
    if (idx >= BB * CC * 64) return;
    int b   = idx >> 14;
    int rem = idx & 16383;
    int c   = rem >> 6;
    int ij  = rem & 63;
    int i   = ij >> 3;
    int j   = ij & 7;
    const float* base = p4 + ((((size_t)b * CC + c) * 32 + i * 4) * 32 + j * 4);
    float s = 0.0f;
    #pragma unroll
    for (int u = 0; u < 4; ++u) {
        #pragma unroll
        for (int v = 0; v < 4; ++v)
            s += base[u * 32 + v];
    }
    X1[idx] = s * (1.0f / 16.0f);
}

// ---------------------------------------------------------------------------
// Bilinear grid-sample (align_corners=False, zeros padding) + coord concat.
// ---------------------------------------------------------------------------
__global__ __launch_bounds__(256)
void dv_sample_kernel(const float* __restrict__ p2,
                      const float* __restrict__ init,
                      float* __restrict__ feat)
{
    const int pt = blockIdx.x;
    const int b  = pt / NPTS;
    const int c  = threadIdx.x;

    const float ix = init[pt * 2 + 0];
    const float iy = init[pt * 2 + 1];
    const float x = ix * 128.0f - 0.5f;
    const float y = iy * 128.0f - 0.5f;
    const float x0f = floorf(x), y0f = floorf(y);
    const float wx1 = x - x0f, wy1 = y - y0f;
    const int x0 = (int)x0f, y0 = (int)y0f;

    const float* __restrict__ img =
        p2 + ((size_t)b * CC + c) * (size_t)(H2W * H2W);

    float acc = 0.0f;
    #pragma unroll
    for (int dy = 0; dy < 2; ++dy) {
        #pragma unroll
        for (int dx = 0; dx < 2; ++dx) {
            const int xi = x0 + dx, yi = y0 + dy;
            const float wx = dx ? wx1 : (1.0f - wx1);
            const float wy = dy ? wy1 : (1.0f - wy1);
            const bool valid = (xi >= 0) && (xi < H2W) && (yi >= 0) && (yi < H2W);
            const int xc = min(max(xi, 0), H2W - 1);
            const int yc = min(max(yi, 0), H2W - 1);
            acc += img[yc * H2W + xc] * (wx * wy * (valid ? 1.0f : 0.0f));
        }
    }

    float* frow = feat + (size_t)pt * FEAT_LD;
    frow[c] = acc;
    if (c == 0) {
        frow[256] = ix;
        frow[257] = iy;
        frow[258] = 0.0f;
        frow[259] = 0.0f;
    }
}

// ---------------------------------------------------------------------------
// disp = tanh(h2 @ R3^T + rb3); refined = clip(init + 0.1*disp, 0, 1).
// ---------------------------------------------------------------------------
__global__ void dv_refine_kernel(const float* __restrict__ h2,
                                 const float* __restrict__ R3,
                                 const float* __restrict__ rb3,
                                 const float* __restrict__ init,
                                 float* __restrict__ refined)
{
    int idx = blockIdx.x * blockDim.x + threadIdx.x;
    if (idx >= BB * NPTS * 2) return;
    const int pt = idx >> 1;
    const int j  = idx & 1;
    const float* __restrict__ hrow = h2 + (size_t)pt * 64;
    const float* __restrict__ wrow = R3 + j * 64;
    float s = rb3[j];
    #pragma unroll 8
    for (int t = 0; t < 64; ++t) s += hrow[t] * wrow[t];
    float r = init[idx] + 0.1f * tanhf(s);
    refined[idx] = fminf(fmaxf(r, 0.0f), 1.0f);
}

// ---------------------------------------------------------------------------
// v = sigmoid(h3 @ V2^T + vb2), h3: [320,128]
// ---------------------------------------------------------------------------
__global__ void dv_score_kernel(const float* __restrict__ h3,
                                const float* __restrict__ V2,
                                const float* __restrict__ vb2,
                                float* __restrict__ vout)
{
    int m = blockIdx.x * blockDim.x + threadIdx.x;
    if (m >= BB * PP) return;
    const float* __restrict__ row = h3 + (size_t)m * 128;
    float s = vb2[0];
    #pragma unroll 8
    for (int j = 0; j < 128; ++j) s += row[j] * V2[j];
    vout[m] = 1.0f / (1.0f + __expf(-s));
}

__global__ void dv_copy_kernel(const float* __restrict__ src,
                               float* __restrict__ dst, int n)
{
    int i = blockIdx.x * blockDim.x + threadIdx.x;
    if (i < n) dst[i] = src[i];
}

// ---------------------------------------------------------------------------
// Launch
// ---------------------------------------------------------------------------
extern "C" void kernel_launch(void* const* d_in, const int* in_sizes, int n_in,
                              void* d_out, int out_size, void* d_ws, size_t ws_size,
                              hipStream_t stream)
{
    (void)in_sizes; (void)n_in; (void)out_size; (void)ws_size;

    const float* p4  = (const float*)d_in[0];
    const float* p2  = (const float*)d_in[1];
    const float* W1  = (const float*)d_in[3];
    const float* b1  = (const float*)d_in[4];
    const float* W2  = (const float*)d_in[5];
    const float* b2  = (const float*)d_in[6];
    const float* R1  = (const float*)d_in[7];
    const float* rb1 = (const float*)d_in[8];
    const float* R2  = (const float*)d_in[9];
    const float* rb2 = (const float*)d_in[10];
    const float* R3  = (const float*)d_in[11];
    const float* rb3 = (const float*)d_in[12];
    const float* V1  = (const float*)d_in[13];
    const float* rvb1 = (const float*)d_in[14];
    const float* V2  = (const float*)d_in[15];
    const float* vb2 = (const float*)d_in[16];

    float* out = (float*)d_out;
    float* out_refined = out;                 // 32000
    float* out_v       = out + 32000;         // 320
    float* out_init    = out + 32320;         // 32000

    float* ws   = (float*)d_ws;
    float* X1   = ws;                              // [16, 16384]
    float* h    = X1   + 262144;                   // [16, 1024]
    float* init = h    + 16384;                    // [16, 2000]
    float* feat = init + 32000;                    // [16000, 260]
    float* h1   = feat + (size_t)16000 * FEAT_LD;  // [16000, 128]
    float* h2w  = h1   + (size_t)16000 * 128;      // [16000, 64]
    float* h3   = h2w  + (size_t)16000 * 64;       // [320, 128]

    // 1) pool
    dv_pool_kernel<<<(BB * CC * 64 + 255) / 256, 256, 0, stream>>>(p4, X1);

    // 2) h = relu(X1 @ W1^T + b1)   M=16, N=1024, K=16384
    dv_wmma_gemm_stream<8><<<dim3(64, 1), dim3(32, 8), 0, stream>>>(
        X1, 16384, W1, 16384, b1, h, 1024, 16384, 1);

    // 3) init = sigmoid(h @ W2^T + b2)  M=16, N=2000, K=1024
    dv_wmma_gemm_stream<8><<<dim3(125, 1), dim3(32, 8), 0, stream>>>(
        h, 1024, W2, 1024, b2, init, 2000, 1024, 2);

    // 4) init_pts -> output
    dv_copy_kernel<<<(32000 + 255) / 256, 256, 0, stream>>>(init, out_init, 32000);

    // 5) sample -> feat [16000, 260]
    dv_sample_kernel<<<BB * NPTS, 256, 0, stream>>>(p2, init, feat);

    // 6) h1 = relu(feat @ R1^T + rb1)  M=16000, N=128, K=258  (TDM-staged W)
    dv_wmma_gemm_lds<8><<<dim3(8, 125), dim3(32, 8), 0, stream>>>(
        feat, FEAT_LD, R1, 258, rb1, h1, 128, 258, 1);

    // 7) h2 = relu(h1 @ R2^T + rb2)    M=16000, N=64, K=128   (TDM-staged W)
    dv_wmma_gemm_lds<8><<<dim3(4, 125), dim3(32, 8), 0, stream>>>(
        h1, 128, R2, 128, rb2, h2w, 64, 128, 1);

    // 8) refined -> output
    dv_refine_kernel<<<(32000 + 255) / 256, 256, 0, stream>>>(
        h2w, R3, rb3, init, out_refined);

    // 9) h3 = relu(refined @ V1^T + vb1)  M=320, N=128, K=100 (TDM-staged W)
    dv_wmma_gemm_lds<4><<<dim3(8, 5), dim3(32, 4), 0, stream>>>(
        out_refined, 100, V1, 100, rvb1, h3, 128, 100, 1);

    // 10) v -> output
    dv_score_kernel<<<(320 + 255) / 256, 256, 0, stream>>>(h3, V2, vb2, out_v);
}